// Baseline_78400333021817
// MI455X (gfx1250) — compile-verified
//
#include <hip/hip_runtime.h>
#include <math.h>

// ---------------- problem constants ----------------
#define Bq   16
#define Tq   2048
#define Lq   512
#define CHq  256
#define HEq  100
#define HDq  200
#define ATq  50
#define VOq  6
#define EMq  64

typedef float v2f __attribute__((ext_vector_type(2)));
typedef float v8f __attribute__((ext_vector_type(8)));

// f32 WMMA, 16x16x4: A = 16x4 (v2f/lane), B = 4x16 (v2f/lane), C/D = 16x16 (v8f/lane)
#define WMMA_F32(a, b, c) \
  __builtin_amdgcn_wmma_f32_16x16x4_f32(false, (a), false, (b), (short)0, (c), false, false)

__device__ __forceinline__ float sigmoidf_(float x) { return 1.0f / (1.0f + __expf(-x)); }

// ---------------------------------------------------------------------------
// Phase 0: collapse the conv res-block. Input has Cin=1, so c1 is rank-1 and
// conv2/conv3 reduce to a 3-tap scalar stencil:
//   Keff[dt][c] = (k1 * k2[dt]) @ k3     (center tap also absorbs conv1a_k)
//   Keff[3][c]  = conv1a_b
// ---------------------------------------------------------------------------
__global__ void k_eff(const float* __restrict__ k1, const float* __restrict__ k1a,
                      const float* __restrict__ b1a, const float* __restrict__ k2,
                      const float* __restrict__ k3, float* __restrict__ Keff) {
  __shared__ float K2[3 * 256];
  int co = threadIdx.x;
  for (int dt = 0; dt < 3; ++dt) {
    float s = 0.f;
    for (int ci = 0; ci < 256; ++ci) s += k1[ci] * k2[(dt * 256 + ci) * 256 + co];
    K2[dt * 256 + co] = s;
  }
  __syncthreads();
  for (int dt = 0; dt < 3; ++dt) {
    float s = 0.f;
    for (int ci = 0; ci < 256; ++ci) s += K2[dt * 256 + ci] * k3[ci * 256 + co];
    if (dt == 1) s += k1a[co];  // fold shortcut conv1a (1x1, Cin=1) into center tap
    Keff[dt * 256 + co] = s;
  }
  Keff[3 * 256 + co] = b1a[co];
}

// feat[b,t,c] = relu(s[t-1]*Ke0 + s[t]*Ke1 + s[t+1]*Ke2 + bias)
__global__ void k_feat(const float* __restrict__ sig, const float* __restrict__ Keff,
                       float* __restrict__ feat) {
  int idx = blockIdx.x * blockDim.x + threadIdx.x;
  if (idx >= Bq * Tq * CHq) return;
  int c = idx & 255, bt = idx >> 8, t = bt & (Tq - 1), b = bt >> 11;
  float sm1 = (t > 0) ? sig[b * Tq + t - 1] : 0.f;
  float s0 = sig[b * Tq + t];
  float sp1 = (t < Tq - 1) ? sig[b * Tq + t + 1] : 0.f;
  float v = sm1 * Keff[c] + s0 * Keff[256 + c] + sp1 * Keff[512 + c] + Keff[768 + c];
  feat[idx] = fmaxf(v, 0.f);
}

// ---------------------------------------------------------------------------
// Encoder input projection: X[b,t,:] = feat[b, tsrc] @ W[0:256,:] + bias
// One wave per 16x16 tile; K=256 -> 64 wmma. Reverse gather folded into A rows.
// ---------------------------------------------------------------------------
__global__ void k_encx(const float* __restrict__ feat, const float* __restrict__ W,
                       const float* __restrict__ bias, const int* __restrict__ len,
                       int reverse, float* __restrict__ X) {
  int wave = threadIdx.x >> 5, lane = threadIdx.x & 31;
  int tile = blockIdx.x * (blockDim.x >> 5) + wave;
  if (tile >= 2048 * 25) return;
  int mt = tile / 25, nt = tile % 25;
  int m = lane & 15, kh = (lane >> 4) * 2, rbase = (lane >> 4) * 8;
  int row = mt * 16 + m;                 // = b*T + t (destination)
  int b = row >> 11, t = row & (Tq - 1);
  int ts = t;
  if (reverse) { int L = len[b]; ts = (t < L) ? (L - 1 - t) : t; }
  const float* arow = feat + (size_t)(b * Tq + ts) * CHq;
  int n = nt * 16 + m;
  v8f acc = {};
  for (int kk = 0; kk < CHq; kk += 4) {
    v2f a;  a.x = arow[kk + kh];            a.y = arow[kk + kh + 1];
    v2f bb; bb.x = W[(kk + kh) * 400 + n];  bb.y = W[(kk + kh + 1) * 400 + n];
    acc = WMMA_F32(a, bb, acc);
  }
  float bn = bias[n];
  for (int r = 0; r < 8; ++r)
    X[(size_t)(mt * 16 + rbase + r) * 400 + n] = acc[r] + bn;
}

// keys = memory @ W_mem  : [32768,200]x[200,200]; N padded to 13 tiles
__global__ void k_keys(const float* __restrict__ mem, const float* __restrict__ Wm,
                       float* __restrict__ keys) {
  int wave = threadIdx.x >> 5, lane = threadIdx.x & 31;
  int tile = blockIdx.x * (blockDim.x >> 5) + wave;
  if (tile >= 2048 * 13) return;
  int mt = tile / 13, nt = tile % 13;
  int m = lane & 15, kh = (lane >> 4) * 2, rbase = (lane >> 4) * 8;
  int n = nt * 16 + m;
  const float* arow = mem + (size_t)(mt * 16 + m) * HDq;
  v8f acc = {};
  for (int kk = 0; kk < HDq; kk += 4) {
    v2f a;  a.x = arow[kk + kh]; a.y = arow[kk + kh + 1];
    v2f bb;
    bb.x = (n < HDq) ? Wm[(kk + kh) * HDq + n] : 0.f;
    bb.y = (n < HDq) ? Wm[(kk + kh + 1) * HDq + n] : 0.f;
    acc = WMMA_F32(a, bb, acc);
  }
  if (n < HDq)
    for (int r = 0; r < 8; ++r)
      keys[(size_t)(mt * 16 + rbase + r) * HDq + n] = acc[r];
}

// decoder embedding projection: Xd[b,l,:] = emb[labels[b,l]] @ W_dec[0:64,:] + b_dec
__global__ void k_decx(const float* __restrict__ embT, const int* __restrict__ labels,
                       const float* __restrict__ Wd, const float* __restrict__ bd,
                       float* __restrict__ Xd) {
  int wave = threadIdx.x >> 5, lane = threadIdx.x & 31;
  int tile = blockIdx.x * (blockDim.x >> 5) + wave;
  if (tile >= 512 * 50) return;
  int mt = tile / 50, nt = tile % 50;
  int m = lane & 15, kh = (lane >> 4) * 2, rbase = (lane >> 4) * 8;
  int row = mt * 16 + m;                 // b*L + l
  const float* arow = embT + (size_t)labels[row] * EMq;
  int n = nt * 16 + m;
  v8f acc = {};
  for (int kk = 0; kk < EMq; kk += 4) {
    v2f a;  a.x = arow[kk + kh];            a.y = arow[kk + kh + 1];
    v2f bb; bb.x = Wd[(kk + kh) * 800 + n]; bb.y = Wd[(kk + kh + 1) * 800 + n];
    acc = WMMA_F32(a, bb, acc);
  }
  float bn = bd[n];
  for (int r = 0; r < 8; ++r)
    Xd[(size_t)(mt * 16 + rbase + r) * 800 + n] = acc[r] + bn;
}

// ---------------------------------------------------------------------------
// Encoder scan: persistent kernel. blockIdx.x = direction (0 fwd, 1 bwd).
// W_h (100x400, 160KB) lives in LDS (320KB/WGP). Each step: 25 WMMA N-tiles
// over 8 waves (K=100 -> 25 wmma each), barrier, gate math, barrier.
// Length masking + output (re)reversal folded into the scatter.
// ---------------------------------------------------------------------------
__global__ void k_enc_scan(const float* __restrict__ Xf, const float* __restrict__ Xb,
                           const float* __restrict__ Wf, const float* __restrict__ Wb,
                           const int* __restrict__ len, float* __restrict__ memory,
                           float* __restrict__ c0h0) {
  extern __shared__ float sm[];
  float* Wh = sm;                 // 100*400
  float* cS = Wh + 40000;         // 16*100
  float* hS = cS + 1600;          // 16*100
  float* zS = hS + 1600;          // 16*400
  int dir = blockIdx.x;
  const float* X = dir ? Xb : Xf;
  const float* W = dir ? Wb : Wf;
  int tid = threadIdx.x;
  for (int i = tid; i < 40000; i += blockDim.x)
    Wh[i] = W[(256 + i / 400) * 400 + (i % 400)];
  for (int i = tid; i < 1600; i += blockDim.x) { cS[i] = 0.f; hS[i] = 0.f; }
  __syncthreads();
  int wave = tid >> 5, lane = tid & 31;
  int m = lane & 15, kh = (lane >> 4) * 2, rbase = (lane >> 4) * 8;
  for (int t = 0; t < Tq; ++t) {
    for (int nt = wave; nt < 25; nt += 8) {
      int n = nt * 16 + m;
      v8f acc = {};
      for (int kk = 0; kk < HEq; kk += 4) {
        v2f a;  a.x = hS[m * HEq + kk + kh];    a.y = hS[m * HEq + kk + kh + 1];
        v2f bb; bb.x = Wh[(kk + kh) * 400 + n]; bb.y = Wh[(kk + kh + 1) * 400 + n];
        acc = WMMA_F32(a, bb, acc);
      }
      for (int r = 0; r < 8; ++r) {
        int b = rbase + r;
        zS[b * 400 + n] = acc[r] + X[(size_t)(b * Tq + t) * 400 + n];
      }
    }
    __syncthreads();
    for (int idx = tid; idx < Bq * HEq; idx += blockDim.x) {
      int b = idx / HEq, j = idx % HEq, L = len[b];
      float zi = zS[b * 400 + j],       zj = zS[b * 400 + 100 + j];
      float zf = zS[b * 400 + 200 + j], zo = zS[b * 400 + 300 + j];
      float nc = cS[idx] * sigmoidf_(zf + 1.f) + sigmoidf_(zi) * tanhf(zj);
      float nh = tanhf(nc) * sigmoidf_(zo);
      if (t < L) {
        cS[idx] = nc; hS[idx] = nh;
        int tout = dir ? (L - 1 - t) : t;
        memory[(size_t)(b * Tq + tout) * HDq + dir * HEq + j] = nh;
      } else {
        memory[(size_t)(b * Tq + t) * HDq + dir * HEq + j] = 0.f;
      }
    }
    __syncthreads();
  }
  for (int idx = tid; idx < Bq * HEq; idx += blockDim.x) {
    int b = idx / HEq, j = idx % HEq;
    c0h0[b * HDq + dir * HEq + j] = cS[idx];
    c0h0[Bq * HDq + b * HDq + dir * HEq + j] = hS[idx];
  }
}

// ---------------------------------------------------------------------------
// Decoder scan: single persistent block, 512 threads (16 waves).
// Per step: WMMA [attn,h](16x250 zero-padded to 252) @ W_dec[64:314] (->800),
// gates, batched score GEMV, masked softmax, context, attention, logits.
// ---------------------------------------------------------------------------
__global__ void k_dec_scan(const float* __restrict__ Xd, const float* __restrict__ Wdec,
                           const float* __restrict__ keys, const float* __restrict__ memory,
                           const float* __restrict__ Wattn, const float* __restrict__ Wout,
                           const float* __restrict__ c0h0, const int* __restrict__ slen,
                           const int* __restrict__ blen, float* __restrict__ out) {
  extern __shared__ float sm[];
  float* zS  = sm;            // 16*800
  float* cS  = zS + 12800;    // 16*200
  float* hS  = cS + 3200;     // 16*200
  float* aS  = hS + 3200;     // 16*50
  float* ahS = aS + 800;      // 16*256 (attn|h|pad)
  float* scS = ahS + 4096;    // 16*2048
  float* cxS = scS + 32768;   // 16*200
  float* red = cxS + 3200;    // 512
  float* inv = red + 512;     // 16
  int tid = threadIdx.x;
  for (int i = tid; i < 3200; i += blockDim.x) { cS[i] = c0h0[i]; hS[i] = c0h0[3200 + i]; }
  for (int i = tid; i < 800; i += blockDim.x) aS[i] = 0.f;
  __syncthreads();
  int wave = tid >> 5, lane = tid & 31;
  int m = lane & 15, kh = (lane >> 4) * 2, rbase = (lane >> 4) * 8;
  const float* Wr = Wdec + 64 * 800;   // rows 64..313 (attn then h)
  for (int l = 0; l < Lq; ++l) {
    for (int i = tid; i < 16 * 256; i += blockDim.x) {
      int b = i >> 8, k = i & 255;
      ahS[i] = (k < 50) ? aS[b * 50 + k] : (k < 250 ? hS[b * HDq + (k - 50)] : 0.f);
    }
    __syncthreads();
    for (int nt = wave; nt < 50; nt += 16) {
      int n = nt * 16 + m;
      v8f acc = {};
      for (int kk = 0; kk < 252; kk += 4) {
        int k0 = kk + kh, k1 = k0 + 1;
        v2f a;  a.x = ahS[m * 256 + k0]; a.y = ahS[m * 256 + k1];
        v2f bb; bb.x = (k0 < 250) ? Wr[k0 * 800 + n] : 0.f;
                bb.y = (k1 < 250) ? Wr[k1 * 800 + n] : 0.f;
        acc = WMMA_F32(a, bb, acc);
      }
      for (int r = 0; r < 8; ++r) {
        int b = rbase + r;
        zS[b * 800 + n] = acc[r] + Xd[(size_t)(b * Lq + l) * 800 + n];
      }
    }
    __syncthreads();
    for (int i = tid; i < Bq * HDq; i += blockDim.x) {
      int b = i / HDq, j = i % HDq;
      float zi = zS[b * 800 + j],       zj = zS[b * 800 + 200 + j];
      float zf = zS[b * 800 + 400 + j], zo = zS[b * 800 + 600 + j];
      float nc = cS[i] * sigmoidf_(zf + 1.f) + sigmoidf_(zi) * tanhf(zj);
      float nh = tanhf(nc) * sigmoidf_(zo);
      cS[i] = nc; hS[i] = nh;
    }
    __syncthreads();
    for (int i = tid; i < Bq * Tq; i += blockDim.x) {
      int b = i >> 11, t = i & (Tq - 1);
      float s = -3.0e38f;
      if (t < slen[b]) {
        const float* kr = keys + (size_t)i * HDq;
        const float* hr = hS + b * HDq;
        float a0 = 0.f;
        for (int d = 0; d < HDq; d += 4)
          a0 += hr[d]*kr[d] + hr[d+1]*kr[d+1] + hr[d+2]*kr[d+2] + hr[d+3]*kr[d+3];
        s = a0;
      }
      scS[i] = s;
    }
    __syncthreads();
    {   // masked softmax; 32 threads per batch row
      int g = tid >> 5, li = tid & 31;
      float lm = -3.4e38f;
      for (int t = li; t < Tq; t += 32) lm = fmaxf(lm, scS[g * Tq + t]);
      red[tid] = lm; __syncthreads();
      if (li == 0) { float mm = -3.4e38f; for (int k = 0; k < 32; ++k) mm = fmaxf(mm, red[(g << 5) + k]); red[g << 5] = mm; }
      __syncthreads();
      float bmax = red[g << 5];
      int L2 = slen[g];
      float ls = 0.f;
      for (int t = li; t < Tq; t += 32) {
        float e = (t < L2) ? __expf(scS[g * Tq + t] - bmax) : 0.f;
        scS[g * Tq + t] = e; ls += e;
      }
      __syncthreads();
      red[tid] = ls; __syncthreads();
      if (li == 0) { float ss = 0.f; for (int k = 0; k < 32; ++k) ss += red[(g << 5) + k]; inv[g] = 1.f / ss; }
    }
    __syncthreads();
    for (int i = tid; i < Bq * HDq; i += blockDim.x) {
      int b = i / HDq, d = i % HDq, L2 = slen[b];
      const float* mb = memory + (size_t)b * Tq * HDq + d;
      const float* sb = scS + b * Tq;
      float s = 0.f;
      for (int t = 0; t < L2; ++t) s += sb[t] * mb[(size_t)t * HDq];
      cxS[i] = s * inv[b];
    }
    __syncthreads();
    for (int i = tid; i < Bq * ATq; i += blockDim.x) {
      int b = i / ATq, a = i % ATq;
      float s = 0.f;
      for (int d = 0; d < HDq; ++d)
        s += hS[b * HDq + d] * Wattn[d * ATq + a] + cxS[b * HDq + d] * Wattn[(HDq + d) * ATq + a];
      aS[i] = s;
    }
    __syncthreads();
    for (int i = tid; i < Bq * VOq; i += blockDim.x) {
      int b = i / VOq, v = i % VOq;
      float s = 0.f;
      for (int a = 0; a < ATq; ++a) s += aS[b * ATq + a] * Wout[a * VOq + v];
      out[(size_t)(b * Lq + l) * VOq + v] = (l < blen[b]) ? s : 0.f;
    }
    __syncthreads();
  }
}

// ---------------------------------------------------------------------------
extern "C" void kernel_launch(void* const* d_in, const int* in_sizes, int n_in,
                              void* d_out, int out_size, void* d_ws, size_t ws_size,
                              hipStream_t stream) {
  const float* sig  = (const float*)d_in[0];
  const float* embT = (const float*)d_in[1];
  const float* k1   = (const float*)d_in[2];
  const float* k1a  = (const float*)d_in[3];
  const float* b1a  = (const float*)d_in[4];
  const float* k2   = (const float*)d_in[5];
  const float* k3   = (const float*)d_in[6];
  const float* Wf   = (const float*)d_in[7];
  const float* bf   = (const float*)d_in[8];
  const float* Wb   = (const float*)d_in[9];
  const float* bbv  = (const float*)d_in[10];
  const float* Wd   = (const float*)d_in[11];
  const float* bd   = (const float*)d_in[12];
  const float* Wm   = (const float*)d_in[13];
  const float* Wa   = (const float*)d_in[14];
  const float* Wo   = (const float*)d_in[15];
  const int* labels = (const int*)d_in[16];
  const int* slen   = (const int*)d_in[17];
  const int* blen   = (const int*)d_in[18];

  float* ws = (float*)d_ws;
  size_t o = 0;
  float* Keff = ws + o; o += 1024;
  float* feat = ws + o; o += (size_t)Bq * Tq * CHq;        // 8.39M
  float* Xf   = ws + o; o += (size_t)Bq * Tq * 400;        // 13.1M
  float* Xb   = ws + o; o += (size_t)Bq * Tq * 400;
  float* mem  = ws + o; o += (size_t)Bq * Tq * HDq;        // 6.55M
  float* keys = ws + o; o += (size_t)Bq * Tq * HDq;
  float* Xd   = ws + o; o += (size_t)Bq * Lq * 800;        // 6.55M
  float* c0h0 = ws + o; o += 2 * Bq * HDq;
  float* outF = (float*)d_out;

  k_eff <<<1, 256, 0, stream>>>(k1, k1a, b1a, k2, k3, Keff);
  k_feat<<<(Bq * Tq * CHq) / 256, 256, 0, stream>>>(sig, Keff, feat);
  k_encx<<<6400, 256, 0, stream>>>(feat, Wf, bf, slen, 0, Xf);
  k_encx<<<6400, 256, 0, stream>>>(feat, Wb, bbv, slen, 1, Xb);
  size_t esm = (size_t)(40000 + 1600 + 1600 + 6400) * sizeof(float);   // ~198 KB LDS
  k_enc_scan<<<2, 256, esm, stream>>>(Xf, Xb, Wf, Wb, slen, mem, c0h0);
  k_keys<<<3328, 256, 0, stream>>>(mem, Wm, keys);
  k_decx<<<3200, 256, 0, stream>>>(embT, labels, Wd, bd, Xd);
  size_t dsm = (size_t)(12800 + 3200 + 3200 + 800 + 4096 + 32768 + 3200 + 512 + 16) * sizeof(float); // ~242 KB
  k_dec_scan<<<1, 512, dsm, stream>>>(Xd, Wd, keys, mem, Wa, Wo, c0h0, slen, blen, outF);
}